// BinarySegmentationLoss_16192026706137
// MI455X (gfx1250) — compile-verified
//
#include <hip/hip_runtime.h>
#include <hip/hip_bf16.h>
#include <math.h>

#define NIMG 8
#define HH   256
#define WW   256
#define BIGF 1e9f
#define EPSF 1e-6f

typedef __attribute__((ext_vector_type(2))) float v2f;
typedef __attribute__((ext_vector_type(8))) float v8f;

// Exact f32 wave32 sum using V_WMMA_F32_16X16X4_F32.
// A layout (16x4 f32): lane m (<16): VGPR0=A[m,0], VGPR1=A[m,1];
//                      lane m+16  : VGPR0=A[m,2], VGPR1=A[m,3].
// We place each lane's value in VGPR0 and zero VGPR1, B = all ones:
//   D[m,n] = v_m + v_{m+16}  (for every column n)
// Per-lane sum of the 8 D VGPRs covers rows {0..7} (lanes 0-15) or {8..15}
// (lanes 16-31); one shfl_xor(16) + add yields the full 32-lane sum in every lane.
__device__ __forceinline__ float wave_sum32(float v) {
    v2f a; a[0] = v;    a[1] = 0.0f;
    v2f b; b[0] = 1.0f; b[1] = 1.0f;
    v8f c = {0.f, 0.f, 0.f, 0.f, 0.f, 0.f, 0.f, 0.f};
    v8f d = __builtin_amdgcn_wmma_f32_16x16x4_f32(
        /*neg_a=*/false, a, /*neg_b=*/false, b,
        /*c_mod=*/(short)0, c, /*reuse_a=*/false, /*reuse_b=*/false);
    float s = d[0] + d[1] + d[2] + d[3] + d[4] + d[5] + d[6] + d[7];
    s += __shfl_xor(s, 16, 32);
    return s;
}

__global__ void k_init_counts(int* __restrict__ fg_count) {
    if (threadIdx.x < NIMG) fg_count[threadIdx.x] = 0;
}

// Column pass: g[y,x] = min_{y'} f[y',x] + (y-y')^2 for fg and bg indicators.
// One block per (image, column); the column lives in LDS. Also counts fg pixels
// per image with integer atomics (deterministic).
__global__ void k_colpass(const float* __restrict__ target,
                          float* __restrict__ g_fg,
                          float* __restrict__ g_bg,
                          int* __restrict__ fg_count) {
    const int n = blockIdx.x / WW;
    const int x = blockIdx.x - n * WW;
    const int y = threadIdx.x;

    __shared__ float f_fg[HH];
    __shared__ float f_bg[HH];
    __shared__ int   scnt;
    if (y == 0) scnt = 0;

    const float t  = target[(n * HH + y) * WW + x];
    const bool  fg = t > 0.5f;
    f_fg[y] = fg ? 0.0f : BIGF;
    f_bg[y] = fg ? BIGF : 0.0f;
    __syncthreads();

    if (fg) atomicAdd(&scnt, 1);

    float gf = BIGF, gb = BIGF;
#pragma unroll 4
    for (int yp = 0; yp < HH; ++yp) {
        const float dy = (float)(y - yp);
        const float d2 = dy * dy;
        gf = fminf(gf, f_fg[yp] + d2);
        gb = fminf(gb, f_bg[yp] + d2);
    }
    g_fg[(n * HH + y) * WW + x] = gf;
    g_bg[(n * HH + y) * WW + x] = gb;

    __syncthreads();
    if (y == 0 && scnt) atomicAdd(&fg_count[n], scnt);
}

// Row pass: d[y,x] = min_{x'} g[y,x'] + (x-x')^2, then phi, sigmoid, and the
// four fused reductions (phi*p, p*t, p, t) via WMMA wave sums -> per-block
// partials (fixed-order combine later; no float atomics).
__global__ void k_rowpass(const float* __restrict__ pred,
                          const float* __restrict__ target,
                          const float* __restrict__ g_fg,
                          const float* __restrict__ g_bg,
                          const int* __restrict__ fg_count,
                          float* __restrict__ partials) {
    const int n    = blockIdx.x / HH;
    const int y    = blockIdx.x - n * HH;
    const int x    = threadIdx.x;
    const int base = (n * HH + y) * WW;

    __shared__ float sf[WW];
    __shared__ float sb[WW];
    __shared__ float wsum[8][4];

    sf[x] = g_fg[base + x];
    sb[x] = g_bg[base + x];
    const float t  = target[base + x];
    const float pr = pred[base + x];
    __syncthreads();

    float df = BIGF, db = BIGF;
#pragma unroll 4
    for (int xp = 0; xp < WW; ++xp) {
        const float dx = (float)(x - xp);
        const float d2 = dx * dx;
        df = fminf(df, sf[xp] + d2);
        db = fminf(db, sb[xp] + d2);
    }
    df = fminf(df, BIGF);
    db = fminf(db, BIGF);
    const float dist_out = sqrtf(df);
    const float dist_in  = sqrtf(db);

    const int   cnt  = fg_count[n];
    const float MAXD = sqrtf((float)((HH - 1) * (HH - 1) + (WW - 1) * (WW - 1)));
    const bool  fg   = t > 0.5f;
    float phi;
    if (cnt == 0)            phi =  MAXD;   // fg empty: dist_outside map = max diag
    else if (cnt == HH * WW) phi = -MAXD;   // bg empty: dist_inside map = max diag
    else                     phi = fg ? -dist_in : dist_out;

    const float p = 1.0f / (1.0f + expf(-pr));

    // WMMA needs EXEC all-ones: uniform control flow here (no divergence above
    // affects participation; all 256 threads reach these calls).
    const float s_b = wave_sum32(phi * p);
    const float s_i = wave_sum32(p * t);
    const float s_p = wave_sum32(p);
    const float s_t = wave_sum32(t);

    const int wave = threadIdx.x >> 5;
    const int lane = threadIdx.x & 31;
    if (lane == 0) {
        wsum[wave][0] = s_b; wsum[wave][1] = s_i;
        wsum[wave][2] = s_p; wsum[wave][3] = s_t;
    }
    __syncthreads();
    if (threadIdx.x == 0) {
        float a0 = 0.f, a1 = 0.f, a2 = 0.f, a3 = 0.f;
        for (int w = 0; w < 8; ++w) {
            a0 += wsum[w][0]; a1 += wsum[w][1];
            a2 += wsum[w][2]; a3 += wsum[w][3];
        }
        float* o = partials + (size_t)blockIdx.x * 4;
        o[0] = a0; o[1] = a1; o[2] = a2; o[3] = a3;
    }
}

// Final combine: wave n reduces the 256 row-partials of image n in a fixed
// schedule, computes per-image dice, then thread 0 emits the scalar loss.
__global__ void k_final(const float* __restrict__ partials,
                        float* __restrict__ out) {
    const int wave = threadIdx.x >> 5;
    const int lane = threadIdx.x & 31;
    __shared__ float dice_sh[NIMG];
    __shared__ float bnd_sh[NIMG];

    float b = 0.f, i = 0.f, p = 0.f, t = 0.f;
    for (int r = lane; r < HH; r += 32) {
        const float* q = partials + (size_t)(wave * HH + r) * 4;
        b += q[0]; i += q[1]; p += q[2]; t += q[3];
    }
    for (int off = 16; off > 0; off >>= 1) {
        b += __shfl_xor(b, off, 32);
        i += __shfl_xor(i, off, 32);
        p += __shfl_xor(p, off, 32);
        t += __shfl_xor(t, off, 32);
    }
    if (lane == 0) {
        dice_sh[wave] = (2.0f * i + EPSF) / (p + t + EPSF);
        bnd_sh[wave]  = b;
    }
    __syncthreads();
    if (threadIdx.x == 0) {
        float dsum = 0.f, bsum = 0.f;
        for (int n = 0; n < NIMG; ++n) { dsum += dice_sh[n]; bsum += bnd_sh[n]; }
        const float dice_loss = 1.0f - dsum / (float)NIMG;
        const float boundary  = bsum / (float)(NIMG * HH * WW);
        out[0] = dice_loss + boundary;
    }
}

extern "C" void kernel_launch(void* const* d_in, const int* in_sizes, int n_in,
                              void* d_out, int out_size, void* d_ws, size_t ws_size,
                              hipStream_t stream) {
    const float* pred   = (const float*)d_in[0];
    const float* target = (const float*)d_in[1];

    float* g_fg     = (float*)d_ws;                                  // 2 MB
    float* g_bg     = g_fg + (size_t)NIMG * HH * WW;                 // 2 MB
    float* partials = g_bg + (size_t)NIMG * HH * WW;                 // 32 KB
    int*   fg_count = (int*)(partials + (size_t)NIMG * HH * 4);      // 32 B

    k_init_counts<<<1, 32, 0, stream>>>(fg_count);
    k_colpass<<<NIMG * WW, HH, 0, stream>>>(target, g_fg, g_bg, fg_count);
    k_rowpass<<<NIMG * HH, WW, 0, stream>>>(pred, target, g_fg, g_bg, fg_count, partials);
    k_final<<<1, NIMG * 32, 0, stream>>>(partials, (float*)d_out);
}